// SISDRLoss_74741020885129
// MI455X (gfx1250) — compile-verified
//
#include <hip/hip_runtime.h>
#include <cstdint>

#define B_ROWS      64
#define T_LEN       131072
#define NCHUNK      32            // chunks per row
#define CHUNK_ELEMS 4096          // elements per chunk per stream
#define TPB         256
#define TILE_ELEMS  1024          // TPB * 4 floats per tile per stream
#define NTILES      4             // CHUNK_ELEMS / TILE_ELEMS

// ---- CDNA5 async global->LDS copy (gfx1250), GVS mode: saddr(SGPR64) + vaddr(VGPR i32) ----
__device__ __forceinline__ void async_load_b128(uint32_t lds_byte_off,
                                                const float* saddr,
                                                uint32_t voff_bytes) {
  asm volatile("global_load_async_to_lds_b128 %0, %1, %2"
               :
               : "v"(lds_byte_off), "v"(voff_bytes), "s"(saddr)
               : "memory");
}
__device__ __forceinline__ void wait_async_le4() {
  asm volatile("s_wait_asynccnt 0x4" ::: "memory");
}
__device__ __forceinline__ void wait_async_0() {
  asm volatile("s_wait_asynccnt 0x0" ::: "memory");
}

__global__ __launch_bounds__(TPB) void sisdr_partial_kernel(
    const float* __restrict__ sd, const float* __restrict__ md,
    const float* __restrict__ lg, const float* __restrict__ tg,
    const int* __restrict__ lens, float* __restrict__ partial) {
  // double buffer: 2 bufs x 4 streams x 1024 floats = 32 KB
  __shared__ __align__(16) float smem[2 * 4 * TILE_ELEMS];
  __shared__ float red[8 * 8];

  const int tid   = threadIdx.x;
  const int row   = (int)(blockIdx.x >> 5);
  const int chunk = (int)(blockIdx.x & (NCHUNK - 1));
  const int L     = lens[row];

  const uint32_t sbase = (uint32_t)(uintptr_t)(&smem[0]);   // low 32 bits = LDS offset
  const uint32_t row_chunk_byte =
      (uint32_t)(row * T_LEN + chunk * CHUNK_ELEMS) * 4u;

  float t2 = 0.f, dss = 0.f, dsm = 0.f, dsl = 0.f;
  float e2s = 0.f, e2m = 0.f, e2l = 0.f;

  auto issue = [&](int tile, int buf) {
    uint32_t goff  = row_chunk_byte + (uint32_t)(tile * TILE_ELEMS + tid * 4) * 4u;
    uint32_t lbase = sbase + (uint32_t)(buf * 4 * TILE_ELEMS + tid * 4) * 4u;
    async_load_b128(lbase + 0u * TILE_ELEMS * 4u, sd, goff);
    async_load_b128(lbase + 1u * TILE_ELEMS * 4u, md, goff);
    async_load_b128(lbase + 2u * TILE_ELEMS * 4u, lg, goff);
    async_load_b128(lbase + 3u * TILE_ELEMS * 4u, tg, goff);
  };

  auto consume = [&](int tile, int buf) {
    const int lidx = buf * 4 * TILE_ELEMS + tid * 4;
    float4 vs = *(const float4*)&smem[lidx + 0 * TILE_ELEMS];
    float4 vm = *(const float4*)&smem[lidx + 1 * TILE_ELEMS];
    float4 vl = *(const float4*)&smem[lidx + 2 * TILE_ELEMS];
    float4 vt = *(const float4*)&smem[lidx + 3 * TILE_ELEMS];
    const int g = chunk * CHUNK_ELEMS + tile * TILE_ELEMS + tid * 4;
    float sv[4] = {vs.x, vs.y, vs.z, vs.w};
    float mv[4] = {vm.x, vm.y, vm.z, vm.w};
    float lv[4] = {vl.x, vl.y, vl.z, vl.w};
    float tv[4] = {vt.x, vt.y, vt.z, vt.w};
#pragma unroll
    for (int j = 0; j < 4; ++j) {
      float msk = ((g + j) < L) ? 1.0f : 0.0f;
      float a = sv[j] * msk, b = mv[j] * msk, c = lv[j] * msk, t = tv[j] * msk;
      t2  = fmaf(t, t, t2);
      dss = fmaf(t, a, dss);
      dsm = fmaf(t, b, dsm);
      dsl = fmaf(t, c, dsl);
      e2s = fmaf(a, a, e2s);
      e2m = fmaf(b, b, e2m);
      e2l = fmaf(c, c, e2l);
    }
  };

  // software pipeline over 4 tiles, depth 2 (ASYNCcnt per-wave, in-order done)
  issue(0, 0);
  issue(1, 1);
  wait_async_le4();  consume(0, 0);  issue(2, 0);
  wait_async_le4();  consume(1, 1);  issue(3, 1);
  wait_async_le4();  consume(2, 0);
  wait_async_0();    consume(3, 1);

  // wave32 shuffle reduction, then cross-wave via LDS
  float acc[7] = {t2, dss, dsm, dsl, e2s, e2m, e2l};
#pragma unroll
  for (int off = 16; off > 0; off >>= 1) {
#pragma unroll
    for (int j = 0; j < 7; ++j) acc[j] += __shfl_down(acc[j], off, 32);
  }
  const int wave = tid >> 5, lane = tid & 31;
  if (lane == 0) {
#pragma unroll
    for (int j = 0; j < 7; ++j) red[wave * 8 + j] = acc[j];
  }
  __syncthreads();
  if (tid == 0) {
    float* p = &partial[(size_t)blockIdx.x * 8];
#pragma unroll
    for (int j = 0; j < 7; ++j) {
      float s = 0.f;
#pragma unroll
      for (int w = 0; w < 8; ++w) s += red[w * 8 + j];
      p[j] = s;
    }
  }
}

__global__ __launch_bounds__(32) void sisdr_row_kernel(
    const float* __restrict__ partial, float* __restrict__ rowloss) {
  const int row = blockIdx.x, lane = threadIdx.x;
  const float* p = &partial[((size_t)row * NCHUNK + lane) * 8];
  float acc[7];
#pragma unroll
  for (int j = 0; j < 7; ++j) acc[j] = p[j];
#pragma unroll
  for (int off = 16; off > 0; off >>= 1) {
#pragma unroll
    for (int j = 0; j < 7; ++j) acc[j] += __shfl_down(acc[j], off, 32);
  }
  if (lane == 0) {
    const float t2  = acc[0];
    const float inv = 1.0f / t2;
    auto sdr = [&](float dot, float e2) {
      float a    = dot * inv;
      float num2 = a * a * t2;                                   // = dot^2 / t2
      float den2 = fmaxf(num2 - 2.0f * a * dot + e2, 0.0f);      // ||a*t - e||^2
      float num  = sqrtf(num2);
      float den  = sqrtf(den2) + 1e-6f;
      return 20.0f * log10f(num / den + 1e-6f);
    };
    float ss = sdr(acc[1], acc[4]);
    float sm = sdr(acc[2], acc[5]);
    float sl = sdr(acc[3], acc[6]);
    rowloss[row] = -(0.8f * ss + 0.1f * sm + 0.1f * sl);
  }
}

__global__ __launch_bounds__(64) void sisdr_final_kernel(
    const float* __restrict__ rowloss, float* __restrict__ out) {
  __shared__ float sh[2];
  const int tid = threadIdx.x;
  float v = rowloss[tid];
#pragma unroll
  for (int off = 16; off > 0; off >>= 1) v += __shfl_down(v, off, 32);
  if ((tid & 31) == 0) sh[tid >> 5] = v;
  __syncthreads();
  if (tid == 0) out[0] = (sh[0] + sh[1]) * (1.0f / (float)B_ROWS);
}

extern "C" void kernel_launch(void* const* d_in, const int* in_sizes, int n_in,
                              void* d_out, int out_size, void* d_ws, size_t ws_size,
                              hipStream_t stream) {
  const float* sd   = (const float*)d_in[0];  // short_decode
  const float* md   = (const float*)d_in[1];  // medium_decode
  const float* lg   = (const float*)d_in[2];  // long_decode
  const float* tg   = (const float*)d_in[3];  // target_audios
  const int*   lens = (const int*)d_in[4];    // target_audios_length

  float* partial = (float*)d_ws;                              // 64*32*8 floats
  float* rowloss = partial + (size_t)B_ROWS * NCHUNK * 8;     // 64 floats

  sisdr_partial_kernel<<<B_ROWS * NCHUNK, TPB, 0, stream>>>(sd, md, lg, tg, lens, partial);
  sisdr_row_kernel<<<B_ROWS, 32, 0, stream>>>(partial, rowloss);
  sisdr_final_kernel<<<1, 64, 0, stream>>>(rowloss, (float*)d_out);
}